// RecurrentEntitiyDecoder_44530220925019
// MI455X (gfx1250) — compile-verified
//
#include <hip/hip_runtime.h>
#include <hip/hip_bf16.h>
#include <cstddef>

// CDNA5 / gfx1250, wave32. FP32 WMMA (V_WMMA_F32_16X16X4_F32).
typedef __attribute__((ext_vector_type(2))) float v2f;
typedef __attribute__((ext_vector_type(8))) float v8f;

#define BATCH 64
#define NENT  128
#define DIM   2048
#define KC    64          // K-chunk staged in LDS
#define LDA   (KC + 4)    // padded LDS row stride (68 floats = 272B, 16B aligned)

// ---------------------------------------------------------------------------
// Kernel 1: fused masked attention -> u[b,d]
// one block per batch row; 8 waves x 32 lanes
// ---------------------------------------------------------------------------
__global__ __launch_bounds__(256) void attn_u_kernel(
    const float* __restrict__ eh,     // [B, E, D]
    const float* __restrict__ q,      // [B, D]
    const unsigned char* __restrict__ mask,  // [B, E] bool
    float* __restrict__ u)            // [B, D]
{
    const int b    = blockIdx.x;
    const int tid  = threadIdx.x;
    const int wave = tid >> 5;
    const int lane = tid & 31;

    __shared__ float s_logit[NENT];
    __shared__ float s_coeff[NENT];

    const float* ehb = eh + (size_t)b * NENT * DIM;
    const float* qb  = q  + (size_t)b * DIM;

    // logits[e] = q . Eh[b,e]  (wave handles 16 entities, lanes stride the dot)
    for (int e = wave; e < NENT; e += 8) {
        const float* row = ehb + (size_t)e * DIM;
        float acc = 0.0f;
        for (int d = lane; d < DIM; d += 32)
            acc += qb[d] * row[d];
        #pragma unroll
        for (int off = 16; off > 0; off >>= 1)
            acc += __shfl_xor(acc, off, 32);
        if (lane == 0)
            s_logit[e] = mask[b * NENT + e] ? acc : -20.0f;
    }
    __syncthreads();

    // stable softmax over 128 entries (redundant per-thread scan; trivial cost)
    float mx = -1e30f;
    for (int e = 0; e < NENT; ++e) mx = fmaxf(mx, s_logit[e]);
    float sum = 0.0f;
    for (int e = 0; e < NENT; ++e) sum += __expf(s_logit[e] - mx);
    const float inv = 1.0f / sum;
    if (tid < NENT)
        s_coeff[tid] = __expf(s_logit[tid] - mx) * inv;
    __syncthreads();

    // u[b,d] = sum_e coeff[e] * Eh[b,e,d]   (coalesced over d)
    for (int d = tid; d < DIM; d += 256) {
        float acc = 0.0f;
        for (int e = 0; e < NENT; ++e)
            acc += s_coeff[e] * ehb[(size_t)e * DIM + d];
        u[(size_t)b * DIM + d] = acc;
    }
}

// ---------------------------------------------------------------------------
// Kernel 2/3: C[64,N] = epilogue(A[64,K] @ Bm[K,N])
//   mode 0: C = A@B + bias           (vocab logits)
//   mode 1: C = sigmoid(addin + A@B) (hidden)
// Each wave owns a full 64-row x 16-col stripe (4 v8f accumulators) so the
// streaming B operand (W_softmax, 412 MB) is read exactly once.
// A chunk staged in LDS, shared by the block's 8 waves.
// B addressing: scalar 64-bit base + 32-bit per-lane element offset (K*N < 2^30
// elements for all our shapes), incremented by 4*N per k-step -> no u64 muls
// in the inner loop.
// ---------------------------------------------------------------------------
__global__ __launch_bounds__(256) void wmma_gemm64(
    const float* __restrict__ A,      // [64, K]
    const float* __restrict__ Bm,     // [K, N]
    float* __restrict__ C,            // [64, N]
    const float* __restrict__ bias,   // [N] or nullptr
    const float* __restrict__ addin,  // [64, N] or nullptr
    int K, int N, int mode)
{
    __shared__ float sA[64 * LDA];

    const int tid  = threadIdx.x;
    const int wave = tid >> 5;
    const int lane = tid & 31;
    const int half = lane >> 4;       // 0: K lo pair, 1: K hi pair (A/B frags)
    const int hl   = lane & 15;       // row (A) / col (B,C) within tile

    const int n0   = (blockIdx.x * 8 + wave) * 16;
    const int ncol = n0 + hl;
    const int ncl  = (ncol < N) ? ncol : (N - 1);   // clamped load column

    const unsigned uN = (unsigned)N;
    // running 32-bit element offset of B[k0+2*half][ncl]
    unsigned boff = (unsigned)(2 * half) * uN + (unsigned)ncl;
    const unsigned bstep = 4u * uN;   // advance per 4-wide k-step

    v8f cacc[4];
    #pragma unroll
    for (int t = 0; t < 4; ++t)
        cacc[t] = (v8f){0.f, 0.f, 0.f, 0.f, 0.f, 0.f, 0.f, 0.f};

    for (int k0 = 0; k0 < K; k0 += KC) {
        __syncthreads();
        // stage A[0..63][k0..k0+KC) -> LDS, coalesced float4 loads
        for (int i = tid; i < 64 * KC / 4; i += 256) {
            const int fidx = i * 4;
            const int row  = fidx >> 6;        // / KC
            const int col  = fidx & (KC - 1);  // % KC
            const float4 v = *(const float4*)(A + (size_t)row * K + k0 + col);
            *(float4*)(sA + row * LDA + col) = v;
        }
        __syncthreads();

        // prefetch next B chunk (global_prefetch_b8)
        if (k0 + KC < K)
            __builtin_prefetch(Bm + (size_t)boff + (size_t)KC * uN, 0, 1);

        #pragma unroll 4
        for (int kk = 0; kk < KC; kk += 4) {
            // B fragment: 4x16 (KxN); lanes 0-15 K={kk,kk+1}, lanes 16-31 K={kk+2,kk+3}
            v2f bfrag;
            bfrag.x = Bm[boff];
            bfrag.y = Bm[boff + uN];
            boff += bstep;
            const int kb = kk + 2 * half;
            #pragma unroll
            for (int t = 0; t < 4; ++t) {
                // A fragment: 16x4 (MxK) from LDS
                const float* Ap = sA + (t * 16 + hl) * LDA + kb;
                v2f afrag;
                afrag.x = Ap[0];
                afrag.y = Ap[1];
                cacc[t] = __builtin_amdgcn_wmma_f32_16x16x4_f32(
                    false, afrag, false, bfrag, (short)0, cacc[t], false, false);
            }
        }
    }

    // epilogue: C/D layout -> VGPR r holds M=r (lanes 0-15) / M=r+8 (lanes 16-31)
    if (ncol < N) {
        const float bv = bias ? bias[ncol] : 0.0f;
        #pragma unroll
        for (int t = 0; t < 4; ++t) {
            #pragma unroll
            for (int r = 0; r < 8; ++r) {
                const int m = t * 16 + half * 8 + r;
                const float acc = cacc[t][r];
                float v;
                if (mode == 1) {
                    const float x = addin[(size_t)m * N + ncol] + acc;
                    v = 1.0f / (1.0f + __expf(-x));
                } else {
                    v = acc + bv;
                }
                C[(size_t)m * N + ncol] = v;
            }
        }
    }
}

// ---------------------------------------------------------------------------
extern "C" void kernel_launch(void* const* d_in, const int* in_sizes, int n_in,
                              void* d_out, int out_size, void* d_ws, size_t ws_size,
                              hipStream_t stream) {
    const float*         eh   = (const float*)d_in[0];          // [64,128,2048]
    const float*         q    = (const float*)d_in[1];          // [64,2048]
    const unsigned char* mask = (const unsigned char*)d_in[2];  // [64,128] bool
    const float*         H    = (const float*)d_in[3];          // [2048,2048]
    const float*         W    = (const float*)d_in[4];          // [2048,V]
    const float*         bsm  = (const float*)d_in[5];          // [V]
    float*               out  = (float*)d_out;                  // [64,V]

    const int V = out_size / BATCH;

    float* u      = (float*)d_ws;          // [64,2048]
    float* hidden = u + BATCH * DIM;       // [64,2048]

    // 1) masked attention + context vector u
    attn_u_kernel<<<BATCH, 256, 0, stream>>>(eh, q, mask, u);

    // 2) hidden = sigmoid(q + u @ H)
    const int tiles2 = DIM / 16;                     // 128 column tiles
    wmma_gemm64<<<(tiles2 + 7) / 8, 256, 0, stream>>>(
        u, H, hidden, nullptr, q, DIM, DIM, 1);

    // 3) out = hidden @ W + b
    const int tiles3 = (V + 15) / 16;
    wmma_gemm64<<<(tiles3 + 7) / 8, 256, 0, stream>>>(
        hidden, W, out, bsm, nullptr, DIM, V, 0);
}